// MessagePassingModule_6305011990992
// MI455X (gfx1250) — compile-verified
//
#include <hip/hip_runtime.h>
#include <stdint.h>

#define WAVES_PER_BLOCK 8
#define THREADS (WAVES_PER_BLOCK * 32)
#define D_FEAT 128
#define DEPTH 4               // async pipeline depth (ring of LDS slots)
#define ROW_BYTES (D_FEAT * 4)  // 512B per edge feature row

// One wave per edge. Lane l handles features [4l, 4l+4).
//  - e rows (512B) staged into LDS via async copies, DEPTH-deep ring (ASYNCcnt)
//    with non-temporal hint so the 256MB stream doesn't evict r/out from L2
//  - r gathers are L2-resident float4 loads
//  - scatter-add via hardware global_atomic_add_f32, device scope
__global__ __launch_bounds__(THREADS)
void mp_scatter_kernel(const float* __restrict__ r,
                       const float* __restrict__ e,
                       const int*   __restrict__ a,
                       float* __restrict__ out,
                       int n_edges, int wave_stride) {
  __shared__ __align__(16) float sbuf[WAVES_PER_BLOCK][DEPTH][D_FEAT];

  const int lane = threadIdx.x & 31;
  const int wib  = threadIdx.x >> 5;
  const int wave = blockIdx.x * WAVES_PER_BLOCK + wib;

  // Low 32 bits of a flat pointer into LDS == LDS byte address (ISA 10.2).
  const uint32_t lds_base = (uint32_t)(uintptr_t)&sbuf[wib][0][lane * 4];

  // ---- prologue: fill up to DEPTH slots -------------------------------------
  int nxt = wave;       // next edge to issue an async load for
  int inflight = 0;     // async loads issued but not yet consumed
#pragma unroll
  for (int k = 0; k < DEPTH; ++k) {
    if (nxt < n_edges) {
      const float*   g   = e + (size_t)nxt * D_FEAT + lane * 4;
      const uint32_t dst = lds_base + (uint32_t)(k * ROW_BYTES);
      asm volatile("global_load_async_to_lds_b128 %0, %1, off th:TH_LOAD_NT"
                   :: "v"(dst), "v"(g) : "memory");
      nxt += wave_stride;
      ++inflight;
    }
  }

  // ---- steady state ---------------------------------------------------------
  int slot = 0;
  for (int cur = wave; cur < n_edges; cur += wave_stride) {
    // Async loads complete in order: asynccnt <= inflight-1 => cur's row ready.
    const int w = inflight - 1;
    if (w >= 3)      asm volatile("s_wait_asynccnt 3" ::: "memory");
    else if (w == 2) asm volatile("s_wait_asynccnt 2" ::: "memory");
    else if (w == 1) asm volatile("s_wait_asynccnt 1" ::: "memory");
    else             asm volatile("s_wait_asynccnt 0" ::: "memory");

    const int2   idx = *(const int2*)(a + 2 * (size_t)cur);   // x=src, y=dst
    const float4 ev  = *(const float4*)&sbuf[wib][slot][lane * 4];
    const float4 rs  = *(const float4*)(r + (size_t)idx.x * D_FEAT + lane * 4);
    const float4 rd  = *(const float4*)(r + (size_t)idx.y * D_FEAT + lane * 4);

    const float4 mij = make_float4(ev.x * rs.x, ev.y * rs.y, ev.z * rs.z, ev.w * rs.w);
    const float4 mji = make_float4(ev.x * rd.x, ev.y * rd.y, ev.z * rd.z, ev.w * rd.w);

    float* pd = out + (size_t)idx.y * D_FEAT + lane * 4;  // src->dst message
    float* ps = out + (size_t)idx.x * D_FEAT + lane * 4;  // dst->src message
    asm volatile(
        "global_atomic_add_f32 %0, %2, off scope:SCOPE_DEV\n\t"
        "global_atomic_add_f32 %0, %3, off offset:4 scope:SCOPE_DEV\n\t"
        "global_atomic_add_f32 %0, %4, off offset:8 scope:SCOPE_DEV\n\t"
        "global_atomic_add_f32 %0, %5, off offset:12 scope:SCOPE_DEV\n\t"
        "global_atomic_add_f32 %1, %6, off scope:SCOPE_DEV\n\t"
        "global_atomic_add_f32 %1, %7, off offset:4 scope:SCOPE_DEV\n\t"
        "global_atomic_add_f32 %1, %8, off offset:8 scope:SCOPE_DEV\n\t"
        "global_atomic_add_f32 %1, %9, off offset:12 scope:SCOPE_DEV"
        :: "v"(pd), "v"(ps),
           "v"(mij.x), "v"(mij.y), "v"(mij.z), "v"(mij.w),
           "v"(mji.x), "v"(mji.y), "v"(mji.z), "v"(mji.w)
        : "memory");

    // The async engine's LDS write is not ordered against this wave's ds_load:
    // drain DScnt before handing the slot back to the async pipeline.
    asm volatile("s_wait_dscnt 0" ::: "memory");
    if (nxt < n_edges) {
      const float*   g   = e + (size_t)nxt * D_FEAT + lane * 4;
      const uint32_t dst = lds_base + (uint32_t)(slot * ROW_BYTES);
      asm volatile("global_load_async_to_lds_b128 %0, %1, off th:TH_LOAD_NT"
                   :: "v"(dst), "v"(g) : "memory");
      nxt += wave_stride;
    } else {
      --inflight;
    }
    slot = (slot + 1) & (DEPTH - 1);
  }
  // s_endpgm performs an implicit wait-idle; outstanding atomics drain there.
}

extern "C" void kernel_launch(void* const* d_in, const int* in_sizes, int n_in,
                              void* d_out, int out_size, void* d_ws, size_t ws_size,
                              hipStream_t stream) {
  (void)n_in; (void)d_ws; (void)ws_size;

  const float* r   = (const float*)d_in[0];   // [N, 128] f32
  const float* e   = (const float*)d_in[1];   // [E, 128] f32
  const int*   a   = (const int*)d_in[2];     // [E, 2] int
  float*       out = (float*)d_out;           // [N, 128] f32

  const int n_edges = in_sizes[2] / 2;

  // Harness poisons d_out; atomics accumulate, so zero it first (capture-safe).
  hipMemsetAsync(d_out, 0, (size_t)out_size * sizeof(float), stream);

  const int blocks      = 2048;
  const int wave_stride = blocks * WAVES_PER_BLOCK;
  mp_scatter_kernel<<<blocks, THREADS, 0, stream>>>(r, e, a, out, n_edges, wave_stride);
}